// TOPKSSMBlockSC_62818191671681
// MI455X (gfx1250) — compile-verified
//
#include <hip/hip_runtime.h>
#include <hip/hip_bf16.h>

// Problem constants (from the reference)
#define BB   16
#define CC   256
#define HH   128
#define WW   128
#define HW   (HH * WW)          // 16384
#define KTOP 2457               // int(16384 * 0.15)

typedef __attribute__((ext_vector_type(2))) float v2f;
typedef __attribute__((ext_vector_type(8))) float v8f;

// ---------------------------------------------------------------------------
// Kernel 1: fused copy x -> out and channel-sum (heat) via chained
// V_WMMA_F32_16X16X4_F32.  One wave handles 16 hw positions x all 256
// channels (64 WMMAs, B = ones).  x is read exactly once.
// Two independent accumulators so adjacent WMMAs have no RAW chain.
//
// A-matrix (16x4 f32) layout per ISA: lanes 0-15 hold (M=lane, K=0..1),
// lanes 16-31 hold (M=lane-16, K=2..3), 2 VGPRs.
// C/D (16x16 f32): lane 0 VGPR r -> (M=r, N=0); lane 16 VGPR r -> (M=8+r, N=0).
// ---------------------------------------------------------------------------
__global__ void __launch_bounds__(256)
heat_copy_kernel(const float* __restrict__ x, float* __restrict__ out,
                 float* __restrict__ heat) {
    const int bid   = blockIdx.x;
    const int b     = bid >> 7;          // / 128
    const int hwblk = bid & 127;         // % 128
    const int tid   = threadIdx.x;
    const int wave  = tid >> 5;
    const int lane  = tid & 31;

    const int hw0 = hwblk * 128 + wave * 16;   // 16 hw positions per wave
    const int m   = lane & 15;                 // row M within tile
    const int kb  = (lane >> 4) << 1;          // K base: 0 or 2

    const size_t plane = (size_t)HW;
    const size_t base  = (size_t)b * CC * plane + (size_t)hw0 + m;
    const float* xb = x + base;
    float*       ob = out + base;

    v8f acc0 = {};                // two independent C accumulators
    v8f acc1 = {};
    v2f ones; ones.x = 1.0f; ones.y = 1.0f;

    #pragma unroll 4
    for (int c0 = 0; c0 < CC; c0 += 8) {
        const size_t o0 = (size_t)(c0 + kb) * plane;
        const size_t o1 = o0 + plane;
        const size_t o2 = o0 + 4 * plane;
        const size_t o3 = o2 + plane;
        float a0 = xb[o0];
        float a1 = xb[o1];
        float a2 = xb[o2];
        float a3 = xb[o3];
        ob[o0] = a0;              // fused copy x -> out
        ob[o1] = a1;
        ob[o2] = a2;
        ob[o3] = a3;
        v2f A0; A0.x = a0; A0.y = a1;
        v2f A1; A1.x = a2; A1.y = a3;
        // D = A(16x4) * ones(4x16) + C : row sums over 4 channels each.
        acc0 = __builtin_amdgcn_wmma_f32_16x16x4_f32(
            false, A0, false, ones, (short)0, acc0, false, false);
        acc1 = __builtin_amdgcn_wmma_f32_16x16x4_f32(
            false, A1, false, ones, (short)0, acc1, false, false);
    }

    // heat = mean over C.  Row M's sum sits in (lane 0, VGPR M) for M<8 and
    // (lane 16, VGPR M-8) for M>=8, column 0.
    const float invC = 1.0f / (float)CC;
    float* hb = heat + (size_t)b * plane + hw0;
    if (lane == 0) {
        #pragma unroll
        for (int r = 0; r < 8; ++r) hb[r] = (acc0[r] + acc1[r]) * invC;
    } else if (lane == 16) {
        #pragma unroll
        for (int r = 0; r < 8; ++r) hb[8 + r] = (acc0[r] + acc1[r]) * invC;
    }
}

// ---------------------------------------------------------------------------
// Kernel 2: exact top-K per batch, output indices sorted ascending.
// One 1024-thread block per batch.  Values cached in registers (16/thread)
// as monotonic sortable uints; 32-step binary search on the bit pattern
// finds the K-th largest value V; ordered compaction (block prefix sums)
// emits indices in ascending order with exact tie handling.
// ---------------------------------------------------------------------------
__device__ __forceinline__ unsigned f2key(float f) {
    unsigned s = __float_as_uint(f);
    return (s & 0x80000000u) ? ~s : (s | 0x80000000u);
}

__global__ void __launch_bounds__(1024)
topk_kernel(const float* __restrict__ heat, int* __restrict__ idx) {
    __shared__ int sh[1024];
    const int b = blockIdx.x;
    const int t = threadIdx.x;
    const float* hb = heat + (size_t)b * HW;

    // cache keys: element index i = chunk*1024 + t
    unsigned u[16];
    #pragma unroll
    for (int j = 0; j < 16; ++j) u[j] = f2key(hb[j * 1024 + t]);

    // --- binary search for V = K-th largest key --------------------------
    unsigned V = 0u;
    for (int bit = 31; bit >= 0; --bit) {
        const unsigned cand = V | (1u << bit);
        int cnt = 0;
        #pragma unroll
        for (int j = 0; j < 16; ++j) cnt += (u[j] >= cand);
        sh[t] = cnt; __syncthreads();
        for (int s = 512; s > 0; s >>= 1) {
            if (t < s) sh[t] += sh[t + s];
            __syncthreads();
        }
        const int total = sh[0];
        __syncthreads();
        if (total >= KTOP) V = cand;
    }

    // --- strictly-greater count -> how many ties to take -----------------
    {
        int cnt = 0;
        #pragma unroll
        for (int j = 0; j < 16; ++j) cnt += (u[j] > V);
        sh[t] = cnt; __syncthreads();
        for (int s = 512; s > 0; s >>= 1) {
            if (t < s) sh[t] += sh[t + s];
            __syncthreads();
        }
    }
    const int count_gt = sh[0];
    __syncthreads();
    const int tie_take = KTOP - count_gt;

    // --- ordered compaction ----------------------------------------------
    int run_eq = 0, run_sel = 0;
    for (int ch = 0; ch < 16; ++ch) {
        const unsigned key = u[ch];
        const int i  = ch * 1024 + t;
        const int eq = (key == V) ? 1 : 0;
        const int gt = (key >  V) ? 1 : 0;

        // inclusive scan of eq
        sh[t] = eq; __syncthreads();
        for (int off = 1; off < 1024; off <<= 1) {
            int o = (t >= off) ? sh[t - off] : 0;
            __syncthreads();
            sh[t] += o;
            __syncthreads();
        }
        const int eq_excl  = sh[t] - eq;
        const int eq_total = sh[1023];
        __syncthreads();

        const int sel = gt | (eq & (((run_eq + eq_excl) < tie_take) ? 1 : 0));

        // inclusive scan of sel
        sh[t] = sel; __syncthreads();
        for (int off = 1; off < 1024; off <<= 1) {
            int o = (t >= off) ? sh[t - off] : 0;
            __syncthreads();
            sh[t] += o;
            __syncthreads();
        }
        const int sel_excl  = sh[t] - sel;
        const int sel_total = sh[1023];
        __syncthreads();

        if (sel) idx[(size_t)b * KTOP + run_sel + sel_excl] = i;
        run_eq  += eq_total;
        run_sel += sel_total;
    }
}

// ---------------------------------------------------------------------------
// Kernel 3: gather + SSM scan + scatter.  One block per batch; thread c
// owns channel c: h_t = sigmoid(a_c)*h_{t-1} + x[b,c,hw_i],
// out[b,c,hw_i] = 2*h_t (row and col branches of the reference are equal).
// Indices staged once in LDS; prefetch a few tokens ahead.
// ---------------------------------------------------------------------------
__global__ void __launch_bounds__(256)
ssm_scatter_kernel(const float* __restrict__ x, const float* __restrict__ a,
                   const int* __restrict__ idx, float* __restrict__ out) {
    __shared__ int sidx[KTOP];
    const int b = blockIdx.x;
    const int c = threadIdx.x;

    for (int i = c; i < KTOP; i += 256) sidx[i] = idx[(size_t)b * KTOP + i];
    __syncthreads();

    const float d = 1.0f / (1.0f + __expf(-a[c]));
    const size_t base = ((size_t)b * CC + c) * (size_t)HW;
    const float* xp = x + base;
    float*       op = out + base;

    float h = 0.0f;
    for (int i = 0; i < KTOP; ++i) {
        const int hw = sidx[i];
        if (i + 8 < KTOP) __builtin_prefetch(xp + sidx[i + 8], 0, 0);
        h = fmaf(d, h, xp[hw]);
        op[hw] = 2.0f * h;
    }
}

// ---------------------------------------------------------------------------
extern "C" void kernel_launch(void* const* d_in, const int* in_sizes, int n_in,
                              void* d_out, int out_size, void* d_ws, size_t ws_size,
                              hipStream_t stream) {
    const float* x = (const float*)d_in[0];   // [B, C, H, W] f32
    const float* a = (const float*)d_in[1];   // [C] f32
    float* out = (float*)d_out;               // [B, C, H, W] f32

    float* heat = (float*)d_ws;                                   // B*HW f32
    int*   idx  = (int*)((char*)d_ws + (size_t)BB * HW * sizeof(float)); // B*KTOP

    // 1) fused copy + channel-mean via WMMA (2048 blocks x 256 threads)
    heat_copy_kernel<<<dim3(BB * (HW / 128)), dim3(256), 0, stream>>>(x, out, heat);
    // 2) exact sorted top-K per batch
    topk_kernel<<<dim3(BB), dim3(1024), 0, stream>>>(heat, idx);
    // 3) SSM scan + scatter
    ssm_scatter_kernel<<<dim3(BB), dim3(256), 0, stream>>>(x, a, idx, out);
}